// SpatialGNNWrapper_30236569764344
// MI455X (gfx1250) — compile-verified
//
#include <hip/hip_runtime.h>
#include <cstdint>
#include <cstddef>

// SAGEConv: out = gelu( mean_agg(x[src]->dst) @ W_l^T + x @ W_r^T + b )
// T=8, N=20000 (divisible by 16), D_IN=D_OUT=128, E=640000.

typedef __attribute__((ext_vector_type(2))) float v2f;
typedef __attribute__((ext_vector_type(4))) float v4f;
typedef __attribute__((ext_vector_type(8))) float v8f;

#define T_STEPS 8
#define D 128

// ---------------- utility: zero a buffer (float4 granularity) ----------------
__global__ void zero_kernel(float4* __restrict__ p, long n4) {
    long i = (long)blockIdx.x * blockDim.x + threadIdx.x;
    if (i < n4) p[i] = make_float4(0.f, 0.f, 0.f, 0.f);
}

// ---------------- in-degree count ----------------
__global__ void deg_kernel(const long long* __restrict__ dst, float* __restrict__ deg, int E) {
    int e = blockIdx.x * blockDim.x + threadIdx.x;
    if (e < E) {
        __hip_atomic_fetch_add(&deg[dst[e]], 1.0f,
                               __ATOMIC_RELAXED, __HIP_MEMORY_SCOPE_AGENT);
    }
}

__global__ void rdeg_kernel(float* __restrict__ deg, int N) {
    int i = blockIdx.x * blockDim.x + threadIdx.x;
    if (i < N) deg[i] = 1.0f / fmaxf(deg[i], 1.0f);
}

// ---------------- pack W_l / W_r into WMMA B-fragment order ----------------
// Bpack layout: [(sel*32 + kstep)*8 + jt][lane][2] floats, so each wave's
// B-fragment load is one coalesced 256B global_load_b64.
// Fragment mapping (V_WMMA_F32_16X16X4_F32 B 4x16): lane l -> N=l%16,
// K pair = kstep*4 + 2*(l/16) + {0,1}; element = W[jt*16 + N][K].
__global__ void pack_w_kernel(const float* __restrict__ Wl,
                              const float* __restrict__ Wr,
                              float* __restrict__ Bpack) {
    int idx = blockIdx.x * blockDim.x + threadIdx.x;  // 2*32*8*32 = 16384
    if (idx >= 16384) return;
    int lane  = idx & 31;
    int jt    = (idx >> 5) & 7;
    int kstep = (idx >> 8) & 31;
    int sel   = (idx >> 13) & 1;
    int n = lane & 15, half = lane >> 4;
    int kk = kstep * 4 + 2 * half;
    const float* W = sel ? Wr : Wl;
    v2f v = *(const v2f*)(W + (jt * 16 + n) * D + kk);
    *(v2f*)(Bpack + (size_t)idx * 2) = v;
}

// ---------------- edge scatter: agg[dst] += x[src] ----------------
// one thread per (edge, 4-float chunk): float4 gather + 4 hw fp32 atomics
__global__ void scatter_kernel(const float* __restrict__ xt,
                               const long long* __restrict__ src,
                               const long long* __restrict__ dst,
                               float* __restrict__ agg, int E) {
    long tid = (long)blockIdx.x * blockDim.x + threadIdx.x;
    long e = tid >> 5;                 // D/4 = 32 chunks per edge
    int  c = (int)(tid & 31) * 4;
    if (e >= E) return;
    long s = (long)src[e];
    long d = (long)dst[e];
    v4f v = *(const v4f*)(xt + s * D + c);
    float* p = agg + d * D + c;
    __hip_atomic_fetch_add(p + 0, v.x, __ATOMIC_RELAXED, __HIP_MEMORY_SCOPE_AGENT);
    __hip_atomic_fetch_add(p + 1, v.y, __ATOMIC_RELAXED, __HIP_MEMORY_SCOPE_AGENT);
    __hip_atomic_fetch_add(p + 2, v.z, __ATOMIC_RELAXED, __HIP_MEMORY_SCOPE_AGENT);
    __hip_atomic_fetch_add(p + 3, v.w, __ATOMIC_RELAXED, __HIP_MEMORY_SCOPE_AGENT);
}

// ---------------- fused WMMA GEMM + bias + exact GELU ----------------
// One wave computes a 16-row x 128-col output strip.
// Phase 1: acc = agg @ W_l^T          (unscaled)
// Scale:   acc[m][:] *= rdeg[m]       (linearity: rdeg is per-row)
// Phase 2: acc += x @ W_r^T
// Epilog:  acc += bias; gelu; store
__global__ __launch_bounds__(256) void sage_gemm_gelu(
    const float* __restrict__ xt,     // [N, D]
    const float* __restrict__ agg,    // [N, D] summed neighbor features
    const float* __restrict__ rdeg,   // [N] 1/max(deg,1)
    const float* __restrict__ Bpack,  // packed W_l|W_r fragments (32768 floats)
    const float* __restrict__ bias,   // [D]
    float* __restrict__ out,          // [N, D]
    int nTiles)                       // N/16
{
    int wave = threadIdx.x >> 5;
    int lane = threadIdx.x & 31;
    int tile = blockIdx.x * 8 + wave;
    if (tile >= nTiles) return;

    int n    = lane & 15;   // A/B fragment row/col within 16-wide tile
    int half = lane >> 4;   // selects K pair {0,1} vs {2,3}
    long row = (long)tile * 16 + n;

    v8f acc[8];
#pragma unroll
    for (int jt = 0; jt < 8; ++jt)
#pragma unroll
        for (int v = 0; v < 8; ++v) acc[jt][v] = 0.f;

    // ---- phase 1: agg @ W_l^T ----
    {
        const float* Arow = agg + row * D;
#pragma unroll 4
        for (int kstep = 0; kstep < 32; ++kstep) {
            v2f a = *(const v2f*)(Arow + kstep * 4 + 2 * half);
            const float* Bp = Bpack + (size_t)(kstep * 8) * 64 + lane * 2;
#pragma unroll
            for (int jt = 0; jt < 8; ++jt) {
                v2f bf = *(const v2f*)(Bp + jt * 64);
                acc[jt] = __builtin_amdgcn_wmma_f32_16x16x4_f32(
                    false, a, false, bf, (short)0, acc[jt], false, false);
            }
        }
    }

    // ---- scale agg partial by 1/deg per OUTPUT row (D row m = v + 8*half) ----
    {
        const float* rp = rdeg + (size_t)tile * 16 + 8 * half;  // 32B aligned
        v4f r0 = *(const v4f*)(rp);
        v4f r1 = *(const v4f*)(rp + 4);
        float rv[8] = {r0.x, r0.y, r0.z, r0.w, r1.x, r1.y, r1.z, r1.w};
#pragma unroll
        for (int jt = 0; jt < 8; ++jt)
#pragma unroll
            for (int v = 0; v < 8; ++v) acc[jt][v] *= rv[v];
    }

    // ---- phase 2: += x @ W_r^T ----
    {
        const float* Arow = xt + row * D;
#pragma unroll 4
        for (int kstep = 0; kstep < 32; ++kstep) {
            v2f a = *(const v2f*)(Arow + kstep * 4 + 2 * half);
            const float* Bp = Bpack + (size_t)((32 + kstep) * 8) * 64 + lane * 2;
#pragma unroll
            for (int jt = 0; jt < 8; ++jt) {
                v2f bf = *(const v2f*)(Bp + jt * 64);
                acc[jt] = __builtin_amdgcn_wmma_f32_16x16x4_f32(
                    false, a, false, bf, (short)0, acc[jt], false, false);
            }
        }
    }

    // ---- bias + exact GELU + store ----
    long rowBase = (long)tile * 16;
#pragma unroll
    for (int jt = 0; jt < 8; ++jt) {
        float bj = bias[jt * 16 + n];
#pragma unroll
        for (int v = 0; v < 8; ++v) {
            float val = acc[jt][v] + bj;
            val = 0.5f * val * (1.0f + erff(val * 0.7071067811865475f));
            out[(rowBase + v + 8 * half) * D + jt * 16 + n] = val;
        }
    }
}

// ---------------- host-side orchestration ----------------
extern "C" void kernel_launch(void* const* d_in, const int* in_sizes, int n_in,
                              void* d_out, int out_size, void* d_ws, size_t ws_size,
                              hipStream_t stream) {
    const float*     x   = (const float*)d_in[0];      // [T, N, D] fp32
    const long long* ei  = (const long long*)d_in[1];  // [2, E] int64
    const float*     Wl  = (const float*)d_in[2];      // [D, D]
    const float*     Wr  = (const float*)d_in[3];      // [D, D]
    const float*     b   = (const float*)d_in[4];      // [D]
    float*           out = (float*)d_out;              // [T, N, D]

    const int E = in_sizes[1] / 2;
    const int N = in_sizes[0] / (T_STEPS * D);
    const long long* src = ei;
    const long long* dst = ei + E;

    // workspace: agg [N*D] | rdeg [N] | Bpack [32768]  (~10.45 MB)
    float* agg   = (float*)d_ws;
    float* rdeg  = agg + (size_t)N * D;
    float* Bpack = rdeg + (size_t)N;

    // ---- once per launch: degree reciprocal + W fragment packing ----
    {
        long n4 = ((long)N + 3) / 4;
        zero_kernel<<<(int)((n4 + 255) / 256), 256, 0, stream>>>((float4*)rdeg, n4);
        deg_kernel<<<(E + 255) / 256, 256, 0, stream>>>(dst, rdeg, E);
        rdeg_kernel<<<(N + 255) / 256, 256, 0, stream>>>(rdeg, N);
        pack_w_kernel<<<16384 / 256, 256, 0, stream>>>(Wl, Wr, Bpack);
    }

    const int nTiles = N / 16;  // N = 20000 divisible by 16
    for (int t = 0; t < T_STEPS; ++t) {
        const float* xt = x + (size_t)t * N * D;
        float* ot       = out + (size_t)t * N * D;

        long n4 = (long)N * D / 4;
        zero_kernel<<<(int)((n4 + 255) / 256), 256, 0, stream>>>((float4*)agg, n4);

        long sth = (long)E * 32;  // one thread per (edge, 4-float chunk)
        scatter_kernel<<<(int)((sth + 255) / 256), 256, 0, stream>>>(xt, src, dst, agg, E);

        sage_gemm_gelu<<<(nTiles + 7) / 8, 256, 0, stream>>>(xt, agg, rdeg, Bpack, b, ot, nTiles);
    }
}